// SAE_36275293782557
// MI455X (gfx1250) — compile-verified
//
#include <hip/hip_runtime.h>

#define HIDDEN 1024
#define LATENT 16384
#define NTOK   8192
#define TOPK   32

// GEMM tiling: C[M,N] = A[M,K] (row-major) x B[N,K]^T (row-major, K contiguous)
#define BM 64
#define BN 128
#define BK 32
#define LDK 36   // padded LDS row stride (floats): conflict-free 16-lane row access,
                 // and 144*row + 16*c4 bytes keeps B128 chunks 16B-aligned.

typedef __attribute__((ext_vector_type(2))) float v2f;
typedef __attribute__((ext_vector_type(8))) float v8f;

// Order-preserving float -> u32 key (larger float => larger key)
__device__ __forceinline__ unsigned fkey(float f) {
  unsigned u = __float_as_uint(f);
  return (u & 0x80000000u) ? ~u : (u | 0x80000000u);
}

// Low 32 bits of a generic LDS pointer == LDS byte offset (flat->LDS truncation).
__device__ __forceinline__ unsigned lds_off(const void* p) {
  return (unsigned)(unsigned long long)(uintptr_t)p;
}

// CDNA5 async memory->LDS copy (16 bytes per lane), tracked by ASYNCcnt.
__device__ __forceinline__ void async_copy_b128(unsigned lds_byte_off, const void* gptr) {
  asm volatile("global_load_async_to_lds_b128 %0, %1, off"
               :: "v"(lds_byte_off), "v"((unsigned long long)(uintptr_t)gptr)
               : "memory");
}

// ---------------------------------------------------------------------------
// Kernel 1: fp32 WMMA GEMM with double-buffered async-to-LDS pipeline.
//   C[m][n] = sum_k A[m][k] * B[n][k]
// A: [M x K] row-major, B: [N x K] row-major (both K-major -> both feed the
// 16x16x4 A/B fragment layout with a contiguous float2 per lane).
// Block: 256 threads = 8 waves; block tile 64x128; wave tile 32x32 (2x2 WMMA).
// Each wave issues 6 async B128 copies per K-tile, one tile ahead of compute.
// ---------------------------------------------------------------------------
__device__ __forceinline__ void stage_tile_async(const float* __restrict__ A,
                                                 const float* __restrict__ B,
                                                 int K, int mBase, int nBase, int k0,
                                                 float* sAbuf, float* sBbuf, int tid) {
#pragma unroll
  for (int i = 0; i < 2; ++i) {           // A tile: 64x32 = 512 float4
    const int idx = tid + i * 256;
    const int row = idx >> 3;
    const int c4  = (idx & 7) << 2;
    async_copy_b128(lds_off(&sAbuf[row * LDK + c4]),
                    &A[(size_t)(mBase + row) * K + k0 + c4]);
  }
#pragma unroll
  for (int i = 0; i < 4; ++i) {           // B tile: 128x32 = 1024 float4
    const int idx = tid + i * 256;
    const int row = idx >> 3;
    const int c4  = (idx & 7) << 2;
    async_copy_b128(lds_off(&sBbuf[row * LDK + c4]),
                    &B[(size_t)(nBase + row) * K + k0 + c4]);
  }
}

__global__ __launch_bounds__(256)
void sae_gemm_f32(const float* __restrict__ A, const float* __restrict__ B,
                  float* __restrict__ C, int M, int N, int K) {
  __shared__ float sA[2][BM * LDK];
  __shared__ float sB[2][BN * LDK];

  const int tid   = threadIdx.x;
  const int mBase = blockIdx.x * BM;
  const int nBase = blockIdx.y * BN;
  const int w     = tid >> 5;
  const int lane  = tid & 31;
  const int wm    = w & 1;     // which 32-row half of the block tile
  const int wn    = w >> 1;    // which 32-col quarter
  const int lr    = lane & 15; // row within 16
  const int lk    = lane >> 4; // K-half select

  const v8f vzero = {0.f, 0.f, 0.f, 0.f, 0.f, 0.f, 0.f, 0.f};
  v8f acc[2][2];
  acc[0][0] = vzero; acc[0][1] = vzero; acc[1][0] = vzero; acc[1][1] = vzero;

  // Prologue: stage tile 0 into buffer 0.
  stage_tile_async(A, B, K, mBase, nBase, 0, sA[0], sB[0], tid);

  int buf = 0;
  for (int k0 = 0; k0 < K; k0 += BK) {
    const int nextk = k0 + BK;
    if (nextk < K) {
      // Issue next tile into the other buffer, then wait until only those 6
      // transfers remain outstanding => current tile's data has landed.
      stage_tile_async(A, B, K, mBase, nBase, nextk, sA[buf ^ 1], sB[buf ^ 1], tid);
      asm volatile("s_wait_asynccnt 0x6" ::: "memory");
    } else {
      asm volatile("s_wait_asynccnt 0x0" ::: "memory");
    }
    __syncthreads();   // all waves' async data for this buffer is visible

    const float* __restrict__ sAb = sA[buf];
    const float* __restrict__ sBb = sB[buf];
#pragma unroll
    for (int kk = 0; kk < BK; kk += 4) {
      const int c = kk + (lk << 1);
      v2f a0 = *(const v2f*)&sAb[(wm * 32 +      lr) * LDK + c];
      v2f a1 = *(const v2f*)&sAb[(wm * 32 + 16 + lr) * LDK + c];
      v2f b0 = *(const v2f*)&sBb[(wn * 32 +      lr) * LDK + c];
      v2f b1 = *(const v2f*)&sBb[(wn * 32 + 16 + lr) * LDK + c];
      acc[0][0] = __builtin_amdgcn_wmma_f32_16x16x4_f32(false, a0, false, b0, (short)0, acc[0][0], false, false);
      acc[0][1] = __builtin_amdgcn_wmma_f32_16x16x4_f32(false, a0, false, b1, (short)0, acc[0][1], false, false);
      acc[1][0] = __builtin_amdgcn_wmma_f32_16x16x4_f32(false, a1, false, b0, (short)0, acc[1][0], false, false);
      acc[1][1] = __builtin_amdgcn_wmma_f32_16x16x4_f32(false, a1, false, b1, (short)0, acc[1][1], false, false);
    }
    __syncthreads();   // WAR: nobody may overwrite this buffer until all done
    buf ^= 1;
  }

  // C/D layout: VGPR v holds row (v + 8*lk), col = lr within the 16x16 tile
#pragma unroll
  for (int mi = 0; mi < 2; ++mi)
#pragma unroll
    for (int ni = 0; ni < 2; ++ni) {
      const int r0 = mBase + wm * 32 + mi * 16 + lk * 8;
      const int cc = nBase + wn * 32 + ni * 16 + lr;
#pragma unroll
      for (int v = 0; v < 8; ++v)
        C[(size_t)(r0 + v) * N + cc] = acc[mi][ni][v];
    }
}

// ---------------------------------------------------------------------------
// Kernel 2: per-row exact top-32 via MSB-first radix select in LDS, then
// in-place scatter+ReLU rewrite of s and compact (idx,val) emission.
// One 256-thread block per token row; the 64KB row lives in LDS.
// ---------------------------------------------------------------------------
__global__ __launch_bounds__(256)
void sae_topk_relu(float* __restrict__ s, int* __restrict__ tIdx,
                   float* __restrict__ tVal, int writeTopk) {
  __shared__ float    rowv[LATENT];
  __shared__ unsigned hist[256];
  __shared__ unsigned sh_bin, sh_need, tieCnt, nsel;
  __shared__ int      tk_i[TOPK];
  __shared__ float    tk_v[TOPK];

  const int tid = threadIdx.x;
  const int n   = blockIdx.x;
  float* rp = s + (size_t)n * LATENT;

  // Load the row once (coalesced float4)
  float4* rv4 = (float4*)rowv;
  const float4* gp4 = (const float4*)rp;
  for (int i = tid; i < LATENT / 4; i += 256) rv4[i] = gp4[i];
  if (tid < TOPK) { tk_i[tid] = 0; tk_v[tid] = 0.f; }
  if (tid == 0)   { tieCnt = 0; nsel = 0; }
  __syncthreads();

  // 4-pass radix select for the exact 32nd-largest key
  unsigned prefix = 0, pmask = 0;
  int need = TOPK;
  for (int shift = 24; shift >= 0; shift -= 8) {
    hist[tid] = 0;
    __syncthreads();
    for (int i = tid; i < LATENT; i += 256) {
      const unsigned k = fkey(rowv[i]);
      if ((k & pmask) == prefix) atomicAdd(&hist[(k >> shift) & 255u], 1u);
    }
    __syncthreads();
    if (tid == 0) {
      int nd = need;
      int b = 255;
      for (; b > 0; --b) {
        const int c = (int)hist[b];
        if (c >= nd) break;
        nd -= c;
      }
      sh_bin  = (unsigned)b;
      sh_need = (unsigned)nd;
    }
    __syncthreads();
    prefix |= sh_bin << shift;
    pmask  |= 0xFFu << shift;
    need    = (int)sh_need;
    __syncthreads();
  }
  const unsigned T = prefix;  // exact threshold key; 'need' copies of T included

  // Rewrite s in place: top-32 keep ReLU(val), rest -> 0. Emit compact lists.
  for (int i = tid; i < LATENT; i += 256) {
    const float v = rowv[i];
    const unsigned k = fkey(v);
    bool sel = (k > T);
    if (!sel && k == T) sel = (atomicAdd(&tieCnt, 1u) < (unsigned)need);
    const float sv = sel ? fmaxf(v, 0.f) : 0.f;
    rp[i] = sv;
    if (sel) {
      const unsigned p = atomicAdd(&nsel, 1u);
      if (p < TOPK) { tk_i[p] = i; tk_v[p] = sv; }
    }
  }
  __syncthreads();
  if (writeTopk && tid < TOPK) {
    tIdx[(size_t)n * TOPK + tid] = tk_i[tid];
    tVal[(size_t)n * TOPK + tid] = tk_v[tid];
  }
}

// ---------------------------------------------------------------------------
// Kernel 3a: transpose W_dec [HIDDEN x LATENT] -> WdT [LATENT x HIDDEN]
// so gather-decode reads contiguous 4KB latent rows (L2-resident, 64MB).
// ---------------------------------------------------------------------------
__global__ __launch_bounds__(256)
void sae_transpose_wdec(const float* __restrict__ Wd, float* __restrict__ WdT) {
  __shared__ float tile[32][33];
  const int l0 = blockIdx.x * 32;
  const int h0 = blockIdx.y * 32;
  const int tx = threadIdx.x & 31;
  const int ty = threadIdx.x >> 5;  // 0..7
#pragma unroll
  for (int r = 0; r < 32; r += 8)
    tile[ty + r][tx] = Wd[(size_t)(h0 + ty + r) * LATENT + l0 + tx];
  __syncthreads();
#pragma unroll
  for (int r = 0; r < 32; r += 8)
    WdT[(size_t)(l0 + ty + r) * HIDDEN + h0 + tx] = tile[tx][ty + r];
}

// ---------------------------------------------------------------------------
// Kernel 3b: sparse gather decode: x_hat[n][:] = sum_j val_j * WdT[idx_j][:]
// 0.54 GFLOP instead of a 275 GFLOP dense GEMM.
// ---------------------------------------------------------------------------
__global__ __launch_bounds__(256)
void sae_decode_gather(const float* __restrict__ WdT, const int* __restrict__ tIdx,
                       const float* __restrict__ tVal, float* __restrict__ xhat) {
  __shared__ int   li[TOPK];
  __shared__ float lv[TOPK];
  const int n   = blockIdx.x;
  const int tid = threadIdx.x;
  if (tid < TOPK) {
    li[tid] = tIdx[(size_t)n * TOPK + tid];
    lv[tid] = tVal[(size_t)n * TOPK + tid];
  }
  __syncthreads();
  const int h = tid << 2;  // 256 threads x float4 = 1024 h
  float4 acc = {0.f, 0.f, 0.f, 0.f};
#pragma unroll 8
  for (int j = 0; j < TOPK; ++j) {
    const float v = lv[j];
    const float4 wv = *(const float4*)&WdT[(size_t)li[j] * HIDDEN + h];
    acc.x += v * wv.x; acc.y += v * wv.y; acc.z += v * wv.z; acc.w += v * wv.w;
  }
  *(float4*)&xhat[(size_t)n * HIDDEN + h] = acc;
}

// ---------------------------------------------------------------------------
extern "C" void kernel_launch(void* const* d_in, const int* in_sizes, int n_in,
                              void* d_out, int out_size, void* d_ws, size_t ws_size,
                              hipStream_t stream) {
  (void)in_sizes; (void)n_in; (void)out_size;
  const float* x  = (const float*)d_in[0];   // [NTOK,  HIDDEN]
  const float* We = (const float*)d_in[1];   // [LATENT,HIDDEN] (K-major for encode)
  const float* Wd = (const float*)d_in[2];   // [HIDDEN,LATENT] (K-major for dense decode)

  float* xhat = (float*)d_out;                         // [NTOK, HIDDEN]
  float* s    = xhat + (size_t)NTOK * HIDDEN;          // [NTOK, LATENT], also holds z

  char* wsb = (char*)d_ws;
  const size_t WDT_BYTES  = (size_t)LATENT * HIDDEN * sizeof(float);  // 64 MB
  const size_t TIDX_BYTES = (size_t)NTOK * TOPK * sizeof(int);        // 1 MB
  const size_t TVAL_BYTES = (size_t)NTOK * TOPK * sizeof(float);      // 1 MB
  const bool gatherPath = ws_size >= WDT_BYTES + TIDX_BYTES + TVAL_BYTES;

  float* WdT  = (float*)wsb;
  int*   tIdx = (int*)(wsb + WDT_BYTES);
  float* tVal = (float*)(wsb + WDT_BYTES + TIDX_BYTES);

  // 1) Encode: z = x * We^T, written into the s region of d_out.
  dim3 gEnc(NTOK / BM, LATENT / BN);
  sae_gemm_f32<<<gEnc, 256, 0, stream>>>(x, We, s, NTOK, LATENT, HIDDEN);

  // 2) Exact per-row top-32 + ReLU, in place; compact lists to workspace.
  sae_topk_relu<<<NTOK, 256, 0, stream>>>(s, tIdx, tVal, gatherPath ? 1 : 0);

  // 3) Decode.
  if (gatherPath) {
    dim3 gT(LATENT / 32, HIDDEN / 32);
    sae_transpose_wdec<<<gT, 256, 0, stream>>>(Wd, WdT);
    sae_decode_gather<<<NTOK, 256, 0, stream>>>(WdT, tIdx, tVal, xhat);
  } else {
    // Workspace too small: dense WMMA decode (W_dec is K-major for this
    // contraction too, so the same GEMM kernel applies unchanged).
    dim3 gDec(NTOK / BM, HIDDEN / BN);
    sae_gemm_f32<<<gDec, 256, 0, stream>>>(s, Wd, xhat, NTOK, HIDDEN, LATENT);
  }
}